// PatchMatchOnce_6158983102662
// MI455X (gfx1250) — compile-verified
//
#include <hip/hip_runtime.h>
#include <hip/hip_bf16.h>

typedef float v2f __attribute__((ext_vector_type(2)));
typedef float v8f __attribute__((ext_vector_type(8)));

#define BB    4
#define CC    256
#define HH    64
#define WW    64
#define HW    4096          // 64*64
#define NCAND 27            // candidates after propagate
#define KTOP  9
#define NSTR  32            // padded candidate stride
#define INVT  100.0f        // 1/TEMPERATURE

// ---------------------------------------------------------------------------
// Kernel 1/5: propagate (h or v) + offset_to_inds.
// offx/offy layout: (b, 9, h, w) == (b, 9, hw). One thread per (b,p,n).
// inds layout: [(b*HW + p)*NSTR + n]
// ---------------------------------------------------------------------------
__global__ void pm_inds_kernel(const float* __restrict__ offx,
                               const float* __restrict__ offy,
                               int* __restrict__ inds,
                               int horizontal) {
    int g = blockIdx.x * blockDim.x + threadIdx.x;   // < 4*27*4096
    int n    = g % NCAND;
    int rest = g / NCAND;                            // b*HW + p
    int p = rest & (HW - 1);
    int b = rest >> 12;
    int x = p & (WW - 1);
    int y = p >> 6;
    int group = n / 9;        // 0: prev-neighbor, 1: self, 2: next-neighbor
    int j     = n % 9;

    float fx = 0.0f, fy = 0.0f;
    if (horizontal) {
        int sx = x + group - 1;
        if (sx >= 0 && sx < WW) {
            size_t o = ((size_t)(b * 9 + j)) * HW + (y << 6) + sx;
            fx = offx[o]; fy = offy[o];
        }
    } else {
        int sy = y + group - 1;
        if (sy >= 0 && sy < HH) {
            size_t o = ((size_t)(b * 9 + j)) * HW + (sy << 6) + x;
            fx = offx[o]; fy = offy[o];
        }
    }
    float cx = fminf(fmaxf((float)x + fx, 0.0f), (float)(WW - 1));
    float cy = fminf(fmaxf((float)y + fy, 0.0f), (float)(HH - 1));
    inds[(size_t)rest * NSTR + n] = (int)(cy * (float)WW + cx);
}

// ---------------------------------------------------------------------------
// Extract diagonal element of a 16x16 f32 D fragment and store cost.
// D layout: VGPR r, lanes 0-15 -> (M=r, N=lane); lanes 16-31 -> (M=8+r, N=lane-16).
// Diag (p,p): p<8 -> lane p, elem p ; p>=8 -> lane p+16 (24..31), elem p-8.
// ---------------------------------------------------------------------------
__device__ __forceinline__ void pm_store_diag(const v8f& acc, int l, int b,
                                              int p0, int n,
                                              float* __restrict__ cost) {
    int e = (l < 8) ? l : (l - 24);
    float dv = acc[0];
    dv = (e == 1) ? acc[1] : dv;
    dv = (e == 2) ? acc[2] : dv;
    dv = (e == 3) ? acc[3] : dv;
    dv = (e == 4) ? acc[4] : dv;
    dv = (e == 5) ? acc[5] : dv;
    dv = (e == 6) ? acc[6] : dv;
    dv = (e == 7) ? acc[7] : dv;
    if (l < 8) {
        cost[((size_t)(b * HW + p0 + l)) * NSTR + n] = dv * INVT;
    } else if (l >= 24) {
        cost[((size_t)(b * HW + p0 + (l - 16))) * NSTR + n] = dv * INVT;
    }
}

// ---------------------------------------------------------------------------
// Kernel 2/6: cost via fp32 WMMA diagonal trick.
// Block = 128 threads = 4 waves, all on one 16-pixel tile (A reuse in L0).
// Wave w handles candidate pairs (n1 = w + 8*i, n2 = n1 + 4): two independent
// WMMA accumulator chains per wave (ILP=2) sharing the A fragment.
// A frag (16x4 f32): lanes 0-15 hold K=0(reg0),K=1(reg1) for M=lane;
//                    lanes 16-31 hold K=2,K=3 for M=lane-16. B mirrored.
// ---------------------------------------------------------------------------
__global__ void pm_cost_wmma_kernel(const float* __restrict__ left,
                                    const float* __restrict__ right,
                                    const int* __restrict__ inds,
                                    float* __restrict__ cost) {
    int tile = blockIdx.x;            // 0..1023  (256 tiles per batch)
    int b    = tile >> 8;
    int p0   = (tile & 255) << 4;     // tile pixel base
    int l    = threadIdx.x & 31;      // lane
    int w    = threadIdx.x >> 5;      // wave 0..3
    int half = l >> 4;                // 0 / 1
    int lp   = l & 15;                // pixel within tile

    const float* Lb = left + (size_t)b * CC * HW;
    const int* indrow = inds + ((size_t)(b * HW + p0 + lp)) * NSTR;

    for (int i = 0; i < 4; ++i) {
        int n1 = w + 8 * i;           // wave-uniform
        if (n1 >= NCAND) break;       // uniform branch: EXEC stays all-ones
        int n2 = n1 + 4;
        bool has2 = (n2 < NCAND);     // wave-uniform

        int col1 = indrow[n1] + b * HW;
        int col2 = indrow[has2 ? n2 : n1] + b * HW;

        v8f acc1 = {0.f, 0.f, 0.f, 0.f, 0.f, 0.f, 0.f, 0.f};
        v8f acc2 = {0.f, 0.f, 0.f, 0.f, 0.f, 0.f, 0.f, 0.f};
        #pragma unroll 4
        for (int k = 0; k < 64; ++k) {
            int kbase = k * 4 + half * 2;
            v2f a, b1, b2;
            a.x  = Lb[(size_t)kbase * HW + p0 + lp];
            a.y  = Lb[(size_t)(kbase + 1) * HW + p0 + lp];
            b1.x = right[(size_t)kbase * (BB * HW) + col1];
            b1.y = right[(size_t)(kbase + 1) * (BB * HW) + col1];
            b2.x = right[(size_t)kbase * (BB * HW) + col2];
            b2.y = right[(size_t)(kbase + 1) * (BB * HW) + col2];
            acc1 = __builtin_amdgcn_wmma_f32_16x16x4_f32(
                false, a, false, b1, (short)0, acc1, false, false);
            acc2 = __builtin_amdgcn_wmma_f32_16x16x4_f32(
                false, a, false, b2, (short)0, acc2, false, false);
        }

        pm_store_diag(acc1, l, b, p0, n1, cost);
        if (has2) pm_store_diag(acc2, l, b, p0, n2, cost);
    }
}

// ---------------------------------------------------------------------------
// Kernel 3/7: softmax normalizers over the 4096 pixels, per (b, n).
// ---------------------------------------------------------------------------
__global__ void pm_softmax_reduce_kernel(const float* __restrict__ cost,
                                         float* __restrict__ mbuf,
                                         float* __restrict__ sbuf) {
    int bn = blockIdx.x;              // 0..4*27-1
    int b = bn / NCAND, n = bn % NCAND;
    __shared__ float red[256];
    int t = threadIdx.x;

    float m = -INFINITY;
    for (int p = t; p < HW; p += 256)
        m = fmaxf(m, cost[((size_t)(b * HW + p)) * NSTR + n]);
    red[t] = m; __syncthreads();
    for (int s = 128; s > 0; s >>= 1) {
        if (t < s) red[t] = fmaxf(red[t], red[t + s]);
        __syncthreads();
    }
    m = red[0]; __syncthreads();

    float sum = 0.0f;
    for (int p = t; p < HW; p += 256)
        sum += expf(cost[((size_t)(b * HW + p)) * NSTR + n] - m);
    red[t] = sum; __syncthreads();
    for (int s = 128; s > 0; s >>= 1) {
        if (t < s) red[t] += red[t + s];
        __syncthreads();
    }
    if (t == 0) { mbuf[b * NSTR + n] = m; sbuf[b * NSTR + n] = red[0]; }
}

// ---------------------------------------------------------------------------
// Kernel 4/8: per-pixel top-9 of softmax values (strict > reproduces JAX
// lowest-index tie-breaking), emit ox/oy (and corr on the final round).
// ox/oy layout: (b, 9, hw)
// ---------------------------------------------------------------------------
__global__ void pm_topk_kernel(const float* __restrict__ cost,
                               const int* __restrict__ inds,
                               const float* __restrict__ mbuf,
                               const float* __restrict__ sbuf,
                               float* __restrict__ ox,
                               float* __restrict__ oy,
                               float* __restrict__ corr) {
    int g = blockIdx.x * blockDim.x + threadIdx.x;   // < 4*4096
    int b = g >> 12;
    int p = g & (HW - 1);
    int x = p & (WW - 1);
    int y = p >> 6;

    float soft[NCAND];
    int   ind[NCAND];
    for (int n = 0; n < NCAND; ++n) {
        float c = cost[(size_t)g * NSTR + n];
        ind[n] = inds[(size_t)g * NSTR + n];
        soft[n] = expf(c - mbuf[b * NSTR + n]) / sbuf[b * NSTR + n];
    }

    unsigned used = 0u;
    for (int j = 0; j < KTOP; ++j) {
        float best = -1.0f; int bi = 0;
        for (int n = 0; n < NCAND; ++n) {
            if (!((used >> n) & 1u) && soft[n] > best) { best = soft[n]; bi = n; }
        }
        used |= 1u << bi;
        int si = ind[bi];
        size_t o = ((size_t)(b * KTOP + j)) * HW + p;
        ox[o] = (float)(si & (WW - 1)) - (float)x;
        oy[o] = (float)(si >> 6) - (float)y;
        if (corr) corr[o] = best;
    }
}

// ---------------------------------------------------------------------------
extern "C" void kernel_launch(void* const* d_in, const int* in_sizes, int n_in,
                              void* d_out, int out_size, void* d_ws, size_t ws_size,
                              hipStream_t stream) {
    const float* left  = (const float*)d_in[0];   // (4, 256, 4096)
    const float* right = (const float*)d_in[1];   // (256, 16384)
    const float* offx  = (const float*)d_in[2];   // (4, 9, 64, 64)
    const float* offy  = (const float*)d_in[3];   // (4, 9, 64, 64)

    // workspace partition (floats / ints)
    float* cost = (float*)d_ws;                       // 4*4096*32   = 524288
    int*   inds = (int*)(cost + BB * HW * NSTR);      // 524288
    float* mbuf = (float*)(inds + BB * HW * NSTR);    // 128
    float* sbuf = mbuf + BB * NSTR;                   // 128
    float* ox1  = sbuf + BB * NSTR;                   // 147456
    float* oy1  = ox1 + BB * KTOP * HW;               // 147456

    float* out_ox   = (float*)d_out;                  // (4,9,64,64)
    float* out_oy   = out_ox + BB * KTOP * HW;
    float* out_corr = out_oy + BB * KTOP * HW;

    const int nIndThreads = BB * NCAND * HW;          // 442368 = 1728*256

    // round 1: horizontal propagate -> evaluate -> intermediate ox/oy
    pm_inds_kernel<<<nIndThreads / 256, 256, 0, stream>>>(offx, offy, inds, 1);
    pm_cost_wmma_kernel<<<BB * HW / 16, 128, 0, stream>>>(left, right, inds, cost);
    pm_softmax_reduce_kernel<<<BB * NCAND, 256, 0, stream>>>(cost, mbuf, sbuf);
    pm_topk_kernel<<<BB * HW / 256, 256, 0, stream>>>(cost, inds, mbuf, sbuf,
                                                      ox1, oy1, nullptr);

    // round 2: vertical propagate -> evaluate -> final outputs
    pm_inds_kernel<<<nIndThreads / 256, 256, 0, stream>>>(ox1, oy1, inds, 0);
    pm_cost_wmma_kernel<<<BB * HW / 16, 128, 0, stream>>>(left, right, inds, cost);
    pm_softmax_reduce_kernel<<<BB * NCAND, 256, 0, stream>>>(cost, mbuf, sbuf);
    pm_topk_kernel<<<BB * HW / 256, 256, 0, stream>>>(cost, inds, mbuf, sbuf,
                                                      out_ox, out_oy, out_corr);
}